// SpGraphAttentionLayer_81552839016630
// MI455X (gfx1250) — compile-verified
//
#include <hip/hip_runtime.h>
#include <hip/hip_bf16.h>
#include <math.h>

typedef __attribute__((ext_vector_type(2))) float v2f;
typedef __attribute__((ext_vector_type(8))) float v8f;

#define IN_DIM   128
#define OUT_DIM  128
#define NR_DIM   64
#define H_DIM    4
#define WROW     320   // 2*IN + NR

// ---------------------------------------------------------------------------
// Kernel 1: fuse M_sel[h] = W_sel[h] @ proj[h]   (sel 0 = src, 1 = tgt)
// grid: x = 64 tiles (8 o-tiles x 8 i-tiles), y = 8 (sel*4 + h); block = 32
// ---------------------------------------------------------------------------
__global__ void fuse_mats_kernel(const float* __restrict__ W,
                                 const float* __restrict__ proj,
                                 float* __restrict__ Mf) {
    const int lane = threadIdx.x;
    const int hi = lane >> 4;       // 0 for lanes 0-15, 1 for lanes 16-31
    const int lm = lane & 15;
    const int tile = blockIdx.x;
    const int o0 = (tile >> 3) * 16;
    const int i0 = (tile & 7) * 16;
    const int sh = blockIdx.y;
    const int sel = sh >> 2;
    const int h = sh & 3;

    const float* Wsel = W + (size_t)h * OUT_DIM * WROW + sel * IN_DIM; // [o][k], stride WROW
    const float* P    = proj + (size_t)h * IN_DIM * IN_DIM;            // [k][i]

    v8f acc = {};
    for (int k0 = 0; k0 < IN_DIM; k0 += 4) {
        const int k = k0 + 2 * hi;
        v2f a; a.x = Wsel[(size_t)(o0 + lm) * WROW + k];
               a.y = Wsel[(size_t)(o0 + lm) * WROW + k + 1];
        v2f b; b.x = P[(size_t)k * IN_DIM + i0 + lm];
               b.y = P[(size_t)(k + 1) * IN_DIM + i0 + lm];
        acc = __builtin_amdgcn_wmma_f32_16x16x4_f32(false, a, false, b,
                                                    (short)0, acc, false, false);
    }
    float* Mout = Mf + (size_t)sh * OUT_DIM * IN_DIM;
#pragma unroll
    for (int v = 0; v < 8; ++v) {
        const int r = v + 8 * hi;
        Mout[(size_t)(o0 + r) * IN_DIM + i0 + lm] = acc[v];
    }
}

// ---------------------------------------------------------------------------
// Kernel 2: hsrc[h] = input @ M_src[h]^T ; htgt[h] = input @ M_tgt[h]^T
// grid: x = N/16, y = 8 (sel*4 + h); block = 32 (one wave per 16x128 strip)
// ---------------------------------------------------------------------------
__global__ void node_gemm_kernel(const float* __restrict__ input,
                                 const float* __restrict__ Mf,
                                 float* __restrict__ hsrc,
                                 float* __restrict__ htgt, int N) {
    const int lane = threadIdx.x;
    const int hi = lane >> 4;
    const int lm = lane & 15;
    const int m0 = blockIdx.x * 16;
    const int sh = blockIdx.y;
    const int sel = sh >> 2;
    const int h = sh & 3;

    const float* M = Mf + (size_t)sh * OUT_DIM * IN_DIM; // [o][i]

    v8f acc[8] = {};
    for (int k0 = 0; k0 < IN_DIM; k0 += 4) {
        const int k = k0 + 2 * hi;
        const float* arow = input + (size_t)(m0 + lm) * IN_DIM + k;
        v2f a; a.x = arow[0]; a.y = arow[1];
#pragma unroll
        for (int nb = 0; nb < 8; ++nb) {
            const int n = nb * 16 + lm;
            v2f b; b.x = M[(size_t)n * IN_DIM + k];
                   b.y = M[(size_t)n * IN_DIM + k + 1];
            acc[nb] = __builtin_amdgcn_wmma_f32_16x16x4_f32(false, a, false, b,
                                                            (short)0, acc[nb], false, false);
        }
    }
    float* dst = (sel == 0 ? hsrc : htgt) + (size_t)h * N * OUT_DIM;
#pragma unroll
    for (int nb = 0; nb < 8; ++nb) {
#pragma unroll
        for (int v = 0; v < 8; ++v) {
            const int r = v + 8 * hi;
            dst[(size_t)(m0 + r) * OUT_DIM + nb * 16 + lm] = acc[nb][v];
        }
    }
}

// ---------------------------------------------------------------------------
// Kernel 3: per-edge phase. One wave handles a 16-edge tile for one head:
//   edge_m tile = (emb @ W_rel^T) via WMMA + gathered hsrc/htgt rows,
//   scores -> edge_e, scatter w[h]*edge_e*edge_m into accOut (atomics),
//   rowsum (last head only).
// Gathered rows are prefetched (global_prefetch_b8) before the WMMA loop so
// HBM latency of the random gathers overlaps the rel-GEMM.
// grid: x = E/16, y = H; block = 32
// ---------------------------------------------------------------------------
__global__ void edge_kernel(const int*   __restrict__ edge,       // (2, E1)
                            const float* __restrict__ edge_embed, // (E1, NR)
                            const int*   __restrict__ edge_nhop,  // (2, E2)
                            const float* __restrict__ emb_nhop,   // (E2, NR)
                            const float* __restrict__ W,          // (H, OUT, 320)
                            const float* __restrict__ a,          // (H, 1, OUT)
                            const float* __restrict__ hw,         // (H,)
                            const float* __restrict__ hsrc,
                            const float* __restrict__ htgt,
                            float* __restrict__ accOut,           // (N, OUT)
                            float* __restrict__ rowsum,           // (N,)
                            int N, int E1, int E2) {
    __shared__ float tile[16 * OUT_DIM];
    __shared__ int   sSrc[16];
    __shared__ int   sTgt[16];
    __shared__ float sEe[16];

    const int lane = threadIdx.x;
    const int hi = lane >> 4;
    const int lm = lane & 15;
    const int h = blockIdx.y;
    const int e0 = blockIdx.x * 16;

    if (lane < 16) {
        const int e = e0 + lane;
        int s, t;
        if (e < E1) { s = edge[e];            t = edge[E1 + e]; }
        else        { s = edge_nhop[e - E1];  t = edge_nhop[E2 + (e - E1)]; }
        sSrc[lane] = s;
        sTgt[lane] = t;
    }
    __syncthreads();

    const float* hs = hsrc + (size_t)h * N * OUT_DIM;
    const float* ht = htgt + (size_t)h * N * OUT_DIM;

    // Prefetch the 32 gathered rows (512 B each = 4 cachelines) while the
    // WMMA rel-GEMM below runs. lanes 0-15 cover hsrc rows, 16-31 htgt rows.
    {
        const float* row = (hi == 0) ? (hs + (size_t)sSrc[lm] * OUT_DIM)
                                     : (ht + (size_t)sTgt[lm] * OUT_DIM);
#pragma unroll
        for (int q = 0; q < 4; ++q)
            __builtin_prefetch(row + q * 32, 0, 3);
    }

    // A-fragment source row for this lane (edge embedding, NR=64 wide)
    const int eRow = e0 + lm;
    const float* embRow = (eRow < E1)
        ? (edge_embed + (size_t)eRow * NR_DIM)
        : (emb_nhop + (size_t)(eRow - E1) * NR_DIM);
    const float* Wrel = W + (size_t)h * OUT_DIM * WROW + 2 * IN_DIM; // [o][r], stride WROW

    v8f acc[8] = {};
    for (int k0 = 0; k0 < NR_DIM; k0 += 4) {
        const int k = k0 + 2 * hi;
        v2f av; av.x = embRow[k]; av.y = embRow[k + 1];
#pragma unroll
        for (int nb = 0; nb < 8; ++nb) {
            const int n = nb * 16 + lm;
            v2f b; b.x = Wrel[(size_t)n * WROW + k];
                   b.y = Wrel[(size_t)n * WROW + k + 1];
            acc[nb] = __builtin_amdgcn_wmma_f32_16x16x4_f32(false, av, false, b,
                                                            (short)0, acc[nb], false, false);
        }
    }

    // add gathered node terms, stage full 16x128 edge_m tile in LDS
#pragma unroll
    for (int v = 0; v < 8; ++v) {
        const int r = v + 8 * hi;
        const int s = sSrc[r];
        const int t = sTgt[r];
#pragma unroll
        for (int nb = 0; nb < 8; ++nb) {
            const int c = nb * 16 + lm;
            tile[r * OUT_DIM + c] = acc[nb][v]
                                  + hs[(size_t)s * OUT_DIM + c]
                                  + ht[(size_t)t * OUT_DIM + c];
        }
    }
    __syncthreads();

    // scores: two lanes per edge, each covering 64 columns
    {
        const float* avec = a + (size_t)h * OUT_DIM;
        float p = 0.f;
        const int cbase = hi * 64;
        for (int c = 0; c < 64; ++c)
            p += tile[lm * OUT_DIM + cbase + c] * avec[cbase + c];
        p += __shfl_xor(p, 16);
        if (lane < 16) {
            const float sc = p;
            const float pw = -(sc > 0.f ? sc : 0.2f * sc);   // -leaky_relu
            sEe[lane] = expf(pw);
        }
    }
    __syncthreads();

    // softmax(head_weights)[h], folded into the scatter
    const float w0 = hw[0], w1 = hw[1], w2 = hw[2], w3 = hw[3];
    const float mx = fmaxf(fmaxf(w0, w1), fmaxf(w2, w3));
    const float es = expf(w0 - mx) + expf(w1 - mx) + expf(w2 - mx) + expf(w3 - mx);
    const float wh = expf(hw[h] - mx) / es;

    // scatter: accOut[src[e], :] += wh * edge_e * edge_m[e, :]
    for (int r = 0; r < 16; ++r) {
        const float coef = wh * sEe[r];
        float* dst = accOut + (size_t)sSrc[r] * OUT_DIM;
#pragma unroll
        for (int c = lane; c < OUT_DIM; c += 32)
            atomicAdd(&dst[c], coef * tile[r * OUT_DIM + c]);
    }
    // rowsum of the LAST head is the divisor in the reference
    if (h == H_DIM - 1 && lane < 16)
        atomicAdd(&rowsum[sSrc[lane]], sEe[lane]);
}

// ---------------------------------------------------------------------------
// Kernel 4: out = elu( accOut / clip(rowsum, 1e-12) )
// ---------------------------------------------------------------------------
__global__ void finalize_kernel(const float* __restrict__ accOut,
                                const float* __restrict__ rowsum,
                                float* __restrict__ out, int total) {
    const int idx = blockIdx.x * blockDim.x + threadIdx.x;
    if (idx < total) {
        const int n = idx >> 7;
        const float rs = fmaxf(rowsum[n], 1e-12f);
        const float x = accOut[idx] / rs;
        out[idx] = x > 0.f ? x : (expf(x) - 1.f);
    }
}

// ---------------------------------------------------------------------------
// Kernel 5: orth_reg = 0.01 * sum_h || proj[h] proj[h]^T - I ||_F
// grid = H blocks of 256 threads; atomicAdd into d_out[N*OUT]
// ---------------------------------------------------------------------------
__global__ void orth_kernel(const float* __restrict__ proj,
                            float* __restrict__ out_scalar) {
    const int h = blockIdx.x;
    const float* P = proj + (size_t)h * IN_DIM * IN_DIM;
    float ssq = 0.f;
    for (int p = threadIdx.x; p < IN_DIM * IN_DIM; p += 256) {
        const int i = p >> 7;
        const int k = p & 127;
        float d = 0.f;
        for (int j = 0; j < IN_DIM; ++j)
            d += P[(size_t)i * IN_DIM + j] * P[(size_t)k * IN_DIM + j];
        if (i == k) d -= 1.f;
        ssq += d * d;
    }
    __shared__ float red[256];
    red[threadIdx.x] = ssq;
    __syncthreads();
    for (int s = 128; s > 0; s >>= 1) {
        if (threadIdx.x < s) red[threadIdx.x] += red[threadIdx.x + s];
        __syncthreads();
    }
    if (threadIdx.x == 0)
        atomicAdd(out_scalar, 0.01f * sqrtf(red[0]));
}

// ---------------------------------------------------------------------------
extern "C" void kernel_launch(void* const* d_in, const int* in_sizes, int n_in,
                              void* d_out, int out_size, void* d_ws, size_t ws_size,
                              hipStream_t stream) {
    const float* input      = (const float*)d_in[0];
    const int*   edge       = (const int*)  d_in[1];
    const float* edge_embed = (const float*)d_in[2];
    const int*   edge_nhop  = (const int*)  d_in[3];
    const float* emb_nhop   = (const float*)d_in[4];
    const float* proj       = (const float*)d_in[5];
    const float* W          = (const float*)d_in[6];
    const float* a          = (const float*)d_in[7];
    const float* hw         = (const float*)d_in[8];

    const int N  = in_sizes[0] / IN_DIM;   // 50000
    const int E1 = in_sizes[1] / 2;        // 200000
    const int E2 = in_sizes[3] / 2;        // 50000
    const int E  = E1 + E2;                // 250000 (multiple of 16)

    float* ws = (float*)d_ws;
    size_t off = 0;
    float* Mf     = ws + off; off += (size_t)2 * H_DIM * OUT_DIM * IN_DIM; // 131072
    float* hsrc   = ws + off; off += (size_t)H_DIM * N * OUT_DIM;          // 25.6M
    float* htgt   = ws + off; off += (size_t)H_DIM * N * OUT_DIM;          // 25.6M
    float* accOut = ws + off; off += (size_t)N * OUT_DIM;                  // 6.4M
    float* rsum   = ws + off; off += (size_t)N;

    float* out = (float*)d_out;
    float* reg_scalar = out + (size_t)N * OUT_DIM;

    hipMemsetAsync(accOut, 0, (size_t)N * OUT_DIM * sizeof(float), stream);
    hipMemsetAsync(rsum,   0, (size_t)N * sizeof(float), stream);
    hipMemsetAsync(reg_scalar, 0, sizeof(float), stream);

    fuse_mats_kernel<<<dim3(64, 8), 32, 0, stream>>>(W, proj, Mf);
    node_gemm_kernel<<<dim3(N / 16, 8), 32, 0, stream>>>(input, Mf, hsrc, htgt, N);
    edge_kernel<<<dim3(E / 16, H_DIM), 32, 0, stream>>>(edge, edge_embed, edge_nhop,
                                                        emb_nhop, W, a, hw,
                                                        hsrc, htgt, accOut, rsum,
                                                        N, E1, E2);
    finalize_kernel<<<dim3((N * OUT_DIM + 255) / 256), 256, 0, stream>>>(accOut, rsum, out,
                                                                         N * OUT_DIM);
    orth_kernel<<<dim3(H_DIM), 256, 0, stream>>>(proj, reg_scalar);
}